// MLDecoder_16415365005966
// MI455X (gfx1250) — compile-verified
//
#include <hip/hip_runtime.h>
#include <hip/hip_bf16.h>
#include <math.h>

typedef __attribute__((ext_vector_type(2))) float v2f;
typedef __attribute__((ext_vector_type(8))) float v8f;

#define B_    64
#define T_    512
#define HID   256
#define NWG_  4

__device__ __forceinline__ v8f wmma_f32(v2f a, v2f b, v8f c) {
  // V_WMMA_F32_16X16X4_F32: D = A(16x4) * B(4x16) + C(16x16), f32 end-to-end
  return __builtin_amdgcn_wmma_f32_16x16x4_f32(false, a, false, b, (short)0, c,
                                               false, false);
}

__device__ __forceinline__ float sigf(float x) { return 1.0f / (1.0f + expf(-x)); }

// CDNA5 async copy: each lane copies 16B from the global address in vaddr to the LDS address in vdst, ASYNCcnt-tracked.
// LDS offset = low 32 bits of the generic pointer (LDS aperture keeps the
// byte offset in addr[31:0]).
__device__ __forceinline__ void async_ld_b128(const float* gaddr, const float* lptr) {
  unsigned loff = (unsigned)(uintptr_t)lptr;
  asm volatile("global_load_async_to_lds_b128 %0, %1, off"
               :: "v"(loff), "v"(gaddr) : "memory");
}
__device__ __forceinline__ void wait_async0() {
  asm volatile("s_wait_asynccnt 0x0" ::: "memory");
}

// ---------------------------------------------------------------------------
// Pair-interleaved transposed weights: Wp[((k>>1)*1024 + n)*2 + (k&1)] = W[n][k]
// -> the WMMA B operand (rows k, k+1 at column n) is one aligned 8B load.
// ---------------------------------------------------------------------------
__global__ void k_init(const float* __restrict__ wih, const float* __restrict__ whh,
                       float* __restrict__ wihTp, float* __restrict__ whhTp,
                       float* __restrict__ hbuf, int* __restrict__ bar) {
  size_t i0 = (size_t)blockIdx.x * blockDim.x + threadIdx.x;
  size_t stride = (size_t)gridDim.x * blockDim.x;
  for (size_t i = i0; i < 819200; i += stride) {   // 800*1024
    int k = (int)(i >> 10), g = (int)(i & 1023);
    wihTp[((((size_t)(k >> 1)) * 1024 + g) << 1) | (k & 1)] = wih[(size_t)g * 800 + k];
  }
  for (size_t i = i0; i < 262144; i += stride) {   // 256*1024
    int k = (int)(i >> 10), g = (int)(i & 1023);
    whhTp[((((size_t)(k >> 1)) * 1024 + g) << 1) | (k & 1)] = whh[(size_t)g * 256 + k];
  }
  for (size_t i = i0; i < 2u * B_ * HID; i += stride) hbuf[i] = 0.0f;
  if (i0 < 2) bar[i0] = 0;
}

// ---------------------------------------------------------------------------
// Kernel 1: pre[t][b][g] = Hi(b,t,:) . Wih[g,32:800] + bih[g] + bhh[g]
// M = 64 batch rows (one t per block), N = 1024, K = 768.
// LDS double-buffered K-panels streamed with global_load_async_to_lds_b128;
// WMMA f32 16x16x4 fed from conflict-free single-b64 LDS reads.
// Block: 256 threads = 8 waves as 2(M) x 4(N); wave tile = 32x32.
// ---------------------------------------------------------------------------
#define KC 32      // K-chunk
#define NCH 24     // 768 / 32
#define BPAD 144   // B tile pair-row stride (floats of pairs): 288 mod 64 = 32
                   // -> half-waves (kpair, kpair+1) hit disjoint bank halves

__global__ __launch_bounds__(256) void k_pre_gemm(
    const float* __restrict__ sents, const float* __restrict__ wihTp,
    const float* __restrict__ bih, const float* __restrict__ bhh,
    float* __restrict__ pre) {
  const int tid   = threadIdx.x;
  const int lane  = tid & 31;
  const int w     = tid >> 5;
  const int wm    = w >> 2;               // 0..1 (M half of block)
  const int wn    = w & 3;                // 0..3 (N quarter of block)
  const int t     = blockIdx.y;           // one timestep per block
  const int nblk  = blockIdx.x * 128;
  const int mrow  = lane & 15;
  const int khalf = (lane >> 4) << 1;     // 0 or 2 (A/B K sub-rows)
  const int rowoff = (lane >> 4) << 3;    // +8 for lanes 16..31 in C/D

  // A rows padded to 36 floats: 16B-aligned rows, 16 distinct banks across
  // the 16 lanes of a half-wave (36*m mod 64 distinct for m=0..15).
  __shared__ __align__(16) float Ast[2][64][36];
  // B tile: [kpair][n][2] pair-interleaved, padded rows.
  __shared__ __align__(16) float Bst[2][KC / 2][BPAD][2];

  // async-copy lane mappings (each element written exactly once per chunk)
  const int a_r0 = w * 8 + (lane >> 3);   // + i*4, i=0..1 -> rows w*8..w*8+7
  const int a_k4 = (lane & 7) * 4;        // 8 lanes x 16B cover a 32-float row
  const int b_q  = w * 4;                 // + j -> 32 copy instrs: 16 kpairs x 2
  const int b_l4 = lane * 4;              // 32 lanes x 16B cover 128 floats

  v8f z = {};
  v8f acc[2][2];
  acc[0][0] = z; acc[0][1] = z; acc[1][0] = z; acc[1][1] = z;

#define ISSUE_CHUNK(c, buf)                                                    \
  {                                                                            \
    const int k0_ = (c) * KC;                                                  \
    _Pragma("unroll")                                                          \
    for (int i = 0; i < 2; i++) {                                              \
      int r = a_r0 + i * 4;                                                    \
      async_ld_b128(sents + ((size_t)r * T_ + t) * 768 + k0_ + a_k4,           \
                    &Ast[buf][r][a_k4]);                                       \
    }                                                                          \
    _Pragma("unroll")                                                          \
    for (int j = 0; j < 4; j++) {                                              \
      int q = b_q + j;                                                         \
      int kpl = q >> 1, half = q & 1;     /* 16 kpair rows x 256 floats */     \
      size_t kp = (size_t)(16 + (k0_ >> 1) + kpl);  /* global kpair row */     \
      async_ld_b128(wihTp + ((kp * 1024 + nblk) << 1) + half * 128 + b_l4,     \
                    &Bst[buf][kpl][0][0] + half * 128 + b_l4);                 \
    }                                                                          \
  }

  ISSUE_CHUNK(0, 0);

  for (int c = 0; c < NCH; c++) {
    const int buf = c & 1;
    wait_async0();        // this wave's copies for chunk c have landed
    __syncthreads();      // all waves' copies for chunk c have landed
    if (c + 1 < NCH) ISSUE_CHUNK(c + 1, buf ^ 1);  // prefetch next chunk

#pragma unroll
    for (int kt = 0; kt < KC; kt += 4) {
      int k = kt + khalf;                 // even
      v2f a0 = *(const v2f*)&Ast[buf][wm * 32 + mrow][k];
      v2f a1 = *(const v2f*)&Ast[buf][wm * 32 + 16 + mrow][k];
#pragma unroll
      for (int ni = 0; ni < 2; ni++) {
        int n = wn * 32 + ni * 16 + mrow;
        v2f bv = *(const v2f*)&Bst[buf][k >> 1][n][0];   // rows k, k+1
        acc[0][ni] = wmma_f32(a0, bv, acc[0][ni]);
        acc[1][ni] = wmma_f32(a1, bv, acc[1][ni]);
      }
    }
    // no trailing barrier needed: the prefetch target buffer was last read
    // in the previous iteration, before this iteration's barrier.
  }

#pragma unroll
  for (int ni = 0; ni < 2; ni++) {
    int n = nblk + wn * 32 + ni * 16 + mrow;
    float bb = bih[n] + bhh[n];
#pragma unroll
    for (int mi = 0; mi < 2; mi++) {
#pragma unroll
      for (int r = 0; r < 8; r++) {
        int brw = wm * 32 + mi * 16 + rowoff + r;   // batch row
        pre[((size_t)t * B_ + brw) * 1024 + n] = acc[mi][ni][r] + bb;
      }
    }
  }
}

// ---------------------------------------------------------------------------
// Kernel 1b: pscore[t][b][tag] = Hi(b,t,:) . Waff[tag,256:1024]
// One wave per (t,b) row; lane-strided K, shuffle reduction.
// ---------------------------------------------------------------------------
__global__ __launch_bounds__(256) void k_pre_score(
    const float* __restrict__ sents, const float* __restrict__ waff,
    float* __restrict__ pscore) {
  const int w = threadIdx.x >> 5, lane = threadIdx.x & 31;
  const int r = blockIdx.x * 8 + w;          // 0..32767
  const int t = r >> 6, b = r & 63;
  const float* x = sents + ((size_t)b * T_ + t) * 768;
  float s0 = 0.0f, s1 = 0.0f;
  for (int k = lane; k < 768; k += 32) {
    float v = x[k];
    s0 += v * waff[256 + k];
    s1 += v * waff[1024 + 256 + k];
  }
  for (int off = 16; off; off >>= 1) {
    s0 += __shfl_xor(s0, off, 32);
    s1 += __shfl_xor(s1, off, 32);
  }
  if (lane == 0) {
    pscore[(size_t)r * 2 + 0] = s0;
    pscore[(size_t)r * 2 + 1] = s1;
  }
}

// ---------------------------------------------------------------------------
// Global sense barrier for the 4 persistent workgroups (one per step).
// ---------------------------------------------------------------------------
__device__ __forceinline__ void grid_barrier(int* cnt, int* gen, int step) {
  __syncthreads();
  if (threadIdx.x == 0) {
    __threadfence();
    int arrived = atomicAdd(cnt, 1);
    if (arrived == NWG_ - 1) {
      atomicExch(cnt, 0);
      __threadfence();
      atomicAdd(gen, 1);
    } else {
      while (__hip_atomic_load(gen, __ATOMIC_ACQUIRE, __HIP_MEMORY_SCOPE_AGENT) <= step) {
        __builtin_amdgcn_s_sleep(2);
      }
    }
    __threadfence();
  }
  __syncthreads();
}

// ---------------------------------------------------------------------------
// Kernel 2: the recurrence. 4 persistent WGs; WG k owns h-cols [64k,64k+64).
// Wave = (batch-half, 16-col slice) and computes all 4 gate types for its
// slice, so i/f/g/o land in matching lane/VGPR slots -> in-register LSTM cell,
// cell state carried in registers across all 512 steps.
// ---------------------------------------------------------------------------
__global__ __launch_bounds__(256) void k_recur(
    const float* __restrict__ mask,
    const float* __restrict__ wihTp, const float* __restrict__ whhTp,
    const float* __restrict__ waff, const float* __restrict__ baff,
    const float* __restrict__ tag_em,
    const float* __restrict__ pre, const float* __restrict__ pscore,
    float* hbuf, float* partial, int* bar_cnt, int* bar_gen,
    float* __restrict__ out) {
  const int wg    = blockIdx.x;
  const int tid   = threadIdx.x;
  const int lane  = tid & 31;
  const int w     = tid >> 5;
  const int mhalf = w & 1;                 // batch half: rows [32*mhalf, +32)
  const int sl    = w >> 1;                // 16-col slice within WG's 64 cols
  const int mrow  = lane & 15;
  const int khalf = (lane >> 4) << 1;
  const int rowoff = (lane >> 4) << 3;
  const int m0    = mhalf * 32;

  __shared__ __align__(16) float yem_lds[64][34];  // even stride: 8B-aligned v2f
  __shared__ float hnew_lds[64][65];
  __shared__ float waff_lds[2][64];
  __shared__ float score_lds[64][2];

  for (int e = tid; e < 64 * 34; e += 256) (&yem_lds[0][0])[e] = 0.0f;  // yem0=0
  if (tid < 128)
    waff_lds[tid >> 6][tid & 63] = waff[(tid >> 6) * 1024 + wg * 64 + (tid & 63)];

  float creg[2][8];                        // cell state, carried in registers
#pragma unroll
  for (int mi = 0; mi < 2; mi++)
#pragma unroll
    for (int r = 0; r < 8; r++) creg[mi][r] = 0.0f;

  int gcol[4];                             // this lane's gate column per type
#pragma unroll
  for (int ty = 0; ty < 4; ty++) gcol[ty] = ty * 256 + wg * 64 + sl * 16 + mrow;
  const int jcol = sl * 16 + mrow;         // col within WG slice
  const int hcol = wg * 64 + jcol;         // global h column
  __syncthreads();

  for (int t = 0; t < T_; ++t) {
    const float* hb = hbuf + (size_t)(t & 1) * (B_ * HID);
    v8f z = {};
    v8f acc[2][4];
#pragma unroll
    for (int mi = 0; mi < 2; mi++)
#pragma unroll
      for (int ty = 0; ty < 4; ty++) acc[mi][ty] = z;

    // --- K part 1: yem (k = 0..31) against WihTp pair-rows 0..15 ---
#pragma unroll
    for (int kt = 0; kt < 32; kt += 4) {
      int k = kt + khalf;                  // even
      v2f a0 = *(const v2f*)&yem_lds[m0 + mrow][k];
      v2f a1 = *(const v2f*)&yem_lds[m0 + 16 + mrow][k];
#pragma unroll
      for (int ty = 0; ty < 4; ty++) {
        v2f bv = *(const v2f*)(wihTp + ((((size_t)(k >> 1)) * 1024 + gcol[ty]) << 1));
        acc[0][ty] = wmma_f32(a0, bv, acc[0][ty]);
        acc[1][ty] = wmma_f32(a1, bv, acc[1][ty]);
      }
    }
    // --- K part 2: h_prev (k = 0..255) against WhhTp ---
    for (int kt = 0; kt < 256; kt += 4) {
      int k = kt + khalf;                  // even
      v2f a0 = *(const v2f*)(hb + (m0 + mrow) * HID + k);
      v2f a1 = *(const v2f*)(hb + (m0 + 16 + mrow) * HID + k);
#pragma unroll
      for (int ty = 0; ty < 4; ty++) {
        v2f bv = *(const v2f*)(whhTp + ((((size_t)(k >> 1)) * 1024 + gcol[ty]) << 1));
        acc[0][ty] = wmma_f32(a0, bv, acc[0][ty]);
        acc[1][ty] = wmma_f32(a1, bv, acc[1][ty]);
      }
    }

    // --- add precomputed input projection, LSTM cell in registers ---
    const float* pr  = pre + (size_t)t * B_ * 1024;
    float*       hbn = hbuf + (size_t)((t + 1) & 1) * (B_ * HID);
#pragma unroll
    for (int mi = 0; mi < 2; mi++) {
#pragma unroll
      for (int r = 0; r < 8; r++) {
        int b = m0 + mi * 16 + rowoff + r;
        float iv = acc[mi][0][r] + pr[(size_t)b * 1024 + gcol[0]];
        float fv = acc[mi][1][r] + pr[(size_t)b * 1024 + gcol[1]];
        float gv = acc[mi][2][r] + pr[(size_t)b * 1024 + gcol[2]];
        float ov = acc[mi][3][r] + pr[(size_t)b * 1024 + gcol[3]];
        float co = creg[mi][r];
        float cn = sigf(fv) * co + sigf(iv) * tanhf(gv);
        float hn = sigf(ov) * tanhf(cn);
        float mv = mask[(size_t)b * T_ + t];
        float hp = hb[b * HID + hcol];
        creg[mi][r] = mv * cn + (1.0f - mv) * co;
        hbn[b * HID + hcol] = mv * hn + (1.0f - mv) * hp;  // carried h (masked)
        hnew_lds[b][jcol] = hn;                            // score uses unmasked
      }
    }
    __syncthreads();

    // --- partial affine scores for this WG's 64 h-cols ---
    if (tid < 128) {
      int b = tid >> 1, tg = tid & 1;
      float sp = 0.0f;
      for (int j = 0; j < 64; j++) sp += hnew_lds[b][j] * waff_lds[tg][j];
      partial[(size_t)((t & 1) * NWG_ + wg) * 128 + tid] = sp;
    }

    grid_barrier(bar_cnt, bar_gen, t);

    // --- reduce partials (redundantly per WG), argmax, tag embedding ---
    if (tid < 128) {
      int b = tid >> 1, tg = tid & 1;
      float sc = baff[tg] + pscore[((size_t)t * B_ + b) * 2 + tg];
      for (int kk = 0; kk < NWG_; kk++)
        sc += partial[(size_t)((t & 1) * NWG_ + kk) * 128 + tid];
      score_lds[b][tg] = sc;
    }
    __syncthreads();
#pragma unroll
    for (int ii = 0; ii < 8; ii++) {
      int e = ii * 256 + tid;              // 64*32 elements
      int b = e >> 5, col = e & 31;
      int pd = score_lds[b][1] > score_lds[b][0] ? 1 : 0;  // argmax, ties -> 0
      yem_lds[b][col] = tag_em[pd * 32 + col];
    }
    if (wg == 0 && tid < 128) {            // log_softmax over 2 tags
      int b = tid >> 1, tg = tid & 1;
      float s0 = score_lds[b][0], s1 = score_lds[b][1];
      float mx = fmaxf(s0, s1);
      float lse = mx + logf(expf(s0 - mx) + expf(s1 - mx));
      out[((size_t)b * T_ + t) * 2 + tg] = score_lds[b][tg] - lse;
    }
    __syncthreads();
  }
}

// ---------------------------------------------------------------------------
extern "C" void kernel_launch(void* const* d_in, const int* in_sizes, int n_in,
                              void* d_out, int out_size, void* d_ws, size_t ws_size,
                              hipStream_t stream) {
  const float* sents  = (const float*)d_in[0];  // [64,512,768]
  const float* mask   = (const float*)d_in[1];  // [64,512]
  const float* wih    = (const float*)d_in[2];  // [1024,800]
  const float* whh    = (const float*)d_in[3];  // [1024,256]
  const float* bih    = (const float*)d_in[4];  // [1024]
  const float* bhh    = (const float*)d_in[5];  // [1024]
  const float* waff   = (const float*)d_in[6];  // [2,1024]
  const float* baff   = (const float*)d_in[7];  // [2]
  const float* tag_em = (const float*)d_in[8];  // [2,32]
  float* out = (float*)d_out;                   // [64,512,2]

  float* ws      = (float*)d_ws;
  int*   bar     = (int*)d_ws;                  // 2 ints in first 256B
  float* wihTp   = ws + 64;                     // 800*1024 (pair-interleaved)
  float* whhTp   = wihTp + 819200;              // 256*1024 (pair-interleaved)
  float* hbuf    = whhTp + 262144;              // 2*64*256
  float* partial = hbuf + 2 * B_ * HID;         // 2*4*128
  float* pscore  = partial + 1024;              // 512*64*2
  float* pre     = pscore + (size_t)T_ * B_ * 2;// 512*64*1024 (134 MB)

  k_init<<<2048, 256, 0, stream>>>(wih, whh, wihTp, whhTp, hbuf, bar);
  dim3 g1(8, T_);                               // N/128 x T
  k_pre_gemm<<<g1, 256, 0, stream>>>(sents, wihTp, bih, bhh, pre);
  k_pre_score<<<4096, 256, 0, stream>>>(sents, waff, pscore);
  k_recur<<<NWG_, 256, 0, stream>>>(mask, wihTp, whhTp, waff, baff, tag_em,
                                    pre, pscore, hbuf, partial,
                                    bar, bar + 1, out);
  (void)in_sizes; (void)n_in; (void)out_size; (void)ws_size;
}